// SocialGNN_34316788695422
// MI455X (gfx1250) — compile-verified
//
#include <hip/hip_runtime.h>
#include <hip/hip_bf16.h>

typedef __attribute__((ext_vector_type(2))) float v2f;
typedef __attribute__((ext_vector_type(8))) float v8f;

// ---------------------------------------------------------------------------
// Degree / normalization
// ---------------------------------------------------------------------------
__global__ void k_deg_init(float* __restrict__ deg, int n) {
    int i = blockIdx.x * blockDim.x + threadIdx.x;
    if (i < n) deg[i] = 1.0f;  // self-loop contributes 1 to every node
}

__global__ void k_deg_edges(const long long* __restrict__ dst,
                            float* __restrict__ deg, long long E) {
    long long e = (long long)blockIdx.x * blockDim.x + threadIdx.x;
    if (e < E) atomicAdd(&deg[dst[e]], 1.0f);
}

__global__ void k_rsqrt_inplace(float* __restrict__ deg, int n) {
    int i = blockIdx.x * blockDim.x + threadIdx.x;
    if (i < n) deg[i] = rsqrtf(deg[i]);  // deg >= 1 always (self-loops)
}

// ---------------------------------------------------------------------------
// GEMM1: H[N x 128] = X[N x 256] * W1[256 x 128], fp32 WMMA 16x16x4.
// Block = 256 threads = 8 waves; wave w computes output tile (blockIdx.x, w).
// A frag: lane = 16*khi + m holds X[row=m][k+2*khi], X[row=m][k+2*khi+1].
// B frag: lane = 16*khi + n holds W[k+2*khi][n],     W[k+2*khi+1][n].
// D:      VGPR r, lane = 16*khi + n -> C[r + 8*khi][n].
// ---------------------------------------------------------------------------
__global__ void k_gemm1_wmma(const float* __restrict__ X,
                             const float* __restrict__ W,
                             float* __restrict__ H) {
    const int lane  = threadIdx.x & 31;
    const int wave  = threadIdx.x >> 5;      // 0..7 -> ntile
    const int mtile = blockIdx.x;            // 0..6249
    const int m     = lane & 15;
    const int khi   = lane >> 4;

    const float* __restrict__ Arow = X + (size_t)(mtile * 16 + m) * 256;
    const float* __restrict__ Bcol = W + wave * 16 + m;   // n == lane&15

    v8f c = {};
#pragma unroll 8
    for (int k = 0; k < 256; k += 4) {
        const int ka = k + 2 * khi;
        v2f a = *reinterpret_cast<const v2f*>(Arow + ka);      // global_load_b64
        v2f b;
        b.x = Bcol[(size_t)ka * 128];
        b.y = Bcol[(size_t)(ka + 1) * 128];
        c = __builtin_amdgcn_wmma_f32_16x16x4_f32(
                /*neg_a=*/false, a, /*neg_b=*/false, b,
                /*c_mod=*/(short)0, c, /*reuse_a=*/false, /*reuse_b=*/false);
    }

    float* __restrict__ out = H + (size_t)(mtile * 16 + 8 * khi) * 128 + wave * 16 + m;
#pragma unroll
    for (int r = 0; r < 8; ++r) out[(size_t)r * 128] = c[r];
}

// ---------------------------------------------------------------------------
// GEMM2: H2[N x 16] = relu(OUT1[N x 128]) * W2[128 x 16]; one wave per mtile.
// ReLU fused into the A-matrix load.
// ---------------------------------------------------------------------------
__global__ void k_gemm2_wmma(const float* __restrict__ A,
                             const float* __restrict__ W,
                             float* __restrict__ H, int mtiles) {
    const int lane  = threadIdx.x & 31;
    const int wave  = threadIdx.x >> 5;
    const int mtile = blockIdx.x * (blockDim.x >> 5) + wave;
    if (mtile >= mtiles) return;             // wave-uniform exit: EXEC stays full
    const int m   = lane & 15;
    const int khi = lane >> 4;

    const float* __restrict__ Arow = A + (size_t)(mtile * 16 + m) * 128;

    v8f c = {};
#pragma unroll 8
    for (int k = 0; k < 128; k += 4) {
        const int ka = k + 2 * khi;
        v2f a = *reinterpret_cast<const v2f*>(Arow + ka);
        a.x = fmaxf(a.x, 0.0f);
        a.y = fmaxf(a.y, 0.0f);
        v2f b;
        b.x = W[ka * 16 + m];
        b.y = W[(ka + 1) * 16 + m];
        c = __builtin_amdgcn_wmma_f32_16x16x4_f32(
                false, a, false, b, (short)0, c, false, false);
    }

    float* __restrict__ out = H + (size_t)(mtile * 16 + 8 * khi) * 16 + m;
#pragma unroll
    for (int r = 0; r < 8; ++r) out[(size_t)r * 16] = c[r];
}

// ---------------------------------------------------------------------------
// Layer-1 aggregation: out1 = b1 + dinv_i^2 * h1_i  (self-loop + bias init),
// then out1[dst] += dinv[src]*dinv[dst] * h1[src] per edge (atomic scatter).
// ---------------------------------------------------------------------------
__global__ void k_init_out1(const float* __restrict__ h1,
                            const float* __restrict__ dinv,
                            const float* __restrict__ b1,
                            float* __restrict__ out1, int n) {
    int t = blockIdx.x * blockDim.x + threadIdx.x;   // n * 32 threads
    int i = t >> 5, lane = t & 31;
    if (i >= n) return;
    float di = dinv[i];
    float w  = di * di;
    float4 hv = reinterpret_cast<const float4*>(h1 + (size_t)i * 128)[lane];
    float4 bv = reinterpret_cast<const float4*>(b1)[lane];
    float4 o;
    o.x = bv.x + w * hv.x; o.y = bv.y + w * hv.y;
    o.z = bv.z + w * hv.z; o.w = bv.w + w * hv.w;
    reinterpret_cast<float4*>(out1 + (size_t)i * 128)[lane] = o;
}

__global__ void k_agg1_edges(const long long* __restrict__ src,
                             const long long* __restrict__ dst,
                             const float* __restrict__ dinv,
                             const float* __restrict__ h1,
                             float* __restrict__ out1, long long E) {
    long long t = (long long)blockIdx.x * blockDim.x + threadIdx.x;
    long long e = t >> 5;                    // 1 wave per edge
    int lane = threadIdx.x & 31;             // 4 channels per lane
    if (e >= E) return;
    long long s = src[e], d = dst[e];
    float norm = dinv[s] * dinv[d];
    float4 hv = reinterpret_cast<const float4*>(h1 + (size_t)s * 128)[lane];
    float* o = out1 + (size_t)d * 128 + 4 * lane;
    atomicAdd(o + 0, norm * hv.x);
    atomicAdd(o + 1, norm * hv.y);
    atomicAdd(o + 2, norm * hv.z);
    atomicAdd(o + 3, norm * hv.w);
}

// ---------------------------------------------------------------------------
// Layer-2 aggregation (16 channels), writing the final output.
// ---------------------------------------------------------------------------
__global__ void k_init_out2(const float* __restrict__ h2,
                            const float* __restrict__ dinv,
                            const float* __restrict__ b2,
                            float* __restrict__ out, int n) {
    int t = blockIdx.x * blockDim.x + threadIdx.x;   // n * 4 threads
    int i = t >> 2, q = t & 3;
    if (i >= n) return;
    float di = dinv[i];
    float w  = di * di;
    float4 hv = reinterpret_cast<const float4*>(h2 + (size_t)i * 16)[q];
    float4 bv = reinterpret_cast<const float4*>(b2)[q];
    float4 o;
    o.x = bv.x + w * hv.x; o.y = bv.y + w * hv.y;
    o.z = bv.z + w * hv.z; o.w = bv.w + w * hv.w;
    reinterpret_cast<float4*>(out + (size_t)i * 16)[q] = o;
}

__global__ void k_agg2_edges(const long long* __restrict__ src,
                             const long long* __restrict__ dst,
                             const float* __restrict__ dinv,
                             const float* __restrict__ h2,
                             float* __restrict__ out, long long E) {
    long long t = (long long)blockIdx.x * blockDim.x + threadIdx.x;
    long long e = t >> 2;                    // 4 threads per edge
    int q = t & 3;
    if (e >= E) return;
    long long s = src[e], d = dst[e];
    float norm = dinv[s] * dinv[d];
    float4 hv = reinterpret_cast<const float4*>(h2 + (size_t)s * 16)[q];
    float* o = out + (size_t)d * 16 + 4 * q;
    atomicAdd(o + 0, norm * hv.x);
    atomicAdd(o + 1, norm * hv.y);
    atomicAdd(o + 2, norm * hv.z);
    atomicAdd(o + 3, norm * hv.w);
}

// ---------------------------------------------------------------------------
static inline int cdiv_i(long long a, long long b) { return (int)((a + b - 1) / b); }

extern "C" void kernel_launch(void* const* d_in, const int* in_sizes, int n_in,
                              void* d_out, int out_size, void* d_ws, size_t ws_size,
                              hipStream_t stream) {
    const float*     x  = (const float*)d_in[0];      // [N, 256]
    const long long* ei = (const long long*)d_in[1];  // [2, E] int64
    const float*     W1 = (const float*)d_in[2];      // [256, 128]
    const float*     b1 = (const float*)d_in[3];      // [128]
    const float*     W2 = (const float*)d_in[4];      // [128, 16]
    const float*     b2 = (const float*)d_in[5];      // [16]
    float*           out = (float*)d_out;             // [N, 16]

    const int       N = in_sizes[0] / 256;            // 100000 (divisible by 16)
    const long long E = (long long)in_sizes[1] / 2;   // 1600000
    const long long* src = ei;
    const long long* dst = ei + E;

    // Workspace layout (256B-aligned): dinv[N] | h1[N*128] | out1[N*128]
    char* ws = (char*)d_ws;
    size_t off_dinv = 0;
    size_t off_h1   = (off_dinv + (size_t)N * 4 + 255) & ~(size_t)255;
    size_t off_out1 = (off_h1 + (size_t)N * 128 * 4 + 255) & ~(size_t)255;
    float* dinv = (float*)(ws + off_dinv);
    float* h1   = (float*)(ws + off_h1);
    float* out1 = (float*)(ws + off_out1);
    float* h2   = h1;  // h1 dead after agg1; reuse its space for [N x 16]

    const int B = 256;
    const int mtiles = N / 16;

    // 1) degrees (self-loop = 1) -> dinv
    k_deg_init<<<cdiv_i(N, B), B, 0, stream>>>(dinv, N);
    k_deg_edges<<<cdiv_i(E, B), B, 0, stream>>>(dst, dinv, E);
    k_rsqrt_inplace<<<cdiv_i(N, B), B, 0, stream>>>(dinv, N);

    // 2) h1 = X @ W1  (fp32 WMMA)
    k_gemm1_wmma<<<mtiles, B, 0, stream>>>(x, W1, h1);

    // 3) out1 = b1 + dinv^2*h1, then edge scatter
    k_init_out1<<<cdiv_i((long long)N * 32, B), B, 0, stream>>>(h1, dinv, b1, out1, N);
    k_agg1_edges<<<cdiv_i(E * 32, B), B, 0, stream>>>(src, dst, dinv, h1, out1, E);

    // 4) h2 = relu(out1) @ W2  (fp32 WMMA, ReLU fused into A-load)
    k_gemm2_wmma<<<cdiv_i(mtiles, B / 32), B, 0, stream>>>(out1, W2, h2, mtiles);

    // 5) out = b2 + dinv^2*h2, then edge scatter
    k_init_out2<<<cdiv_i((long long)N * 4, B), B, 0, stream>>>(h2, dinv, b2, out, N);
    k_agg2_edges<<<cdiv_i(E * 4, B), B, 0, stream>>>(src, dst, dinv, h2, out, E);
}